// DTVNet_60610578482024
// MI455X (gfx1250) — compile-verified
//
#include <hip/hip_runtime.h>
#include <hip/hip_bf16.h>

// ---------------------------------------------------------------------------
// DTV-Net forward for MI455X (gfx1250, wave32).
// Volume: [1,1,64,128,128] fp32.  Convs: 3x3x3, CH=4, mapped to
// V_WMMA_F32_16X16X4_F32 (K = CH = 4, M = 16 x-voxels, N = Cout cols).
// ---------------------------------------------------------------------------

#define Dz 64
#define Hy 128
#define Wx 128
#define HW  (Hy * Wx)            // 16384
#define VOL (Dz * Hy * Wx)       // 1048576
#define NCASC 3

typedef float v2f __attribute__((ext_vector_type(2)));
typedef float v8f __attribute__((ext_vector_type(8)));

// ---------------------------------------------------------------------------
// Elementwise / stencil kernels
// ---------------------------------------------------------------------------

__global__ void copy_k(float* __restrict__ dst, const float* __restrict__ src, int n) {
    int i = blockIdx.x * blockDim.x + threadIdx.x;
    if (i < n) dst[i] = src[i];
}

__global__ void zero_k(float* __restrict__ dst, int n) {
    int i = blockIdx.x * blockDim.x + threadIdx.x;
    if (i < n) dst[i] = 0.0f;
}

// resid[h,w] = sino[h,w] - sum_d t[d,h,w]
__global__ void resid_k(const float* __restrict__ t, const float* __restrict__ sino,
                        float* __restrict__ resid) {
    int i = blockIdx.x * blockDim.x + threadIdx.x;
    if (i >= HW) return;
    float s = 0.0f;
    for (int d = 0; d < Dz; ++d) s += t[d * HW + i];
    resid[i] = sino[i] - s;
}

// z = t + resid/Dz (broadcast along depth)
__global__ void z_k(const float* __restrict__ t, const float* __restrict__ resid,
                    float* __restrict__ z) {
    int i = blockIdx.x * blockDim.x + threadIdx.x;
    if (i >= VOL) return;
    z[i] = t[i] + resid[i & (HW - 1)] * (1.0f / (float)Dz);
}

// forward differences along x, y, z (last slice = 0)
__global__ void diff_k(const float* __restrict__ z, float* __restrict__ pn,
                       float* __restrict__ qn, float* __restrict__ sn) {
    int i = blockIdx.x * blockDim.x + threadIdx.x;
    if (i >= VOL) return;
    int x = i & (Wx - 1);
    int y = (i >> 7) & (Hy - 1);
    int d = i >> 14;
    float zc = z[i];
    pn[i] = (x < Wx - 1) ? z[i + 1]  - zc : 0.0f;
    qn[i] = (y < Hy - 1) ? z[i + Wx] - zc : 0.0f;
    sn[i] = (d < Dz - 1) ? z[i + HW] - zc : 0.0f;
}

// p = p + nt*(p - (blk - pn))
__global__ void upd_k(float* __restrict__ p, const float* __restrict__ blk,
                      const float* __restrict__ pn, const float* __restrict__ ntv, int c) {
    int i = blockIdx.x * blockDim.x + threadIdx.x;
    if (i >= VOL) return;
    float nt = ntv[c];
    float p_ = blk[i] - pn[i];
    float pv = p[i];
    p[i] = pv + nt * (pv - p_);
}

// t_next = Dx^T p + Dy^T q + Dz^T s + t + nt*(t - znew)
__global__ void combine_k(const float* __restrict__ t, const float* __restrict__ p,
                          const float* __restrict__ q, const float* __restrict__ s,
                          const float* __restrict__ znew, const float* __restrict__ ntv,
                          int c, float* __restrict__ tn) {
    int i = blockIdx.x * blockDim.x + threadIdx.x;
    if (i >= VOL) return;
    int x = i & (Wx - 1);
    int y = (i >> 7) & (Hy - 1);
    int d = i >> 14;
    float tx = ((x > 0) ? p[i - 1]  : 0.0f) - ((x < Wx - 1) ? p[i] : 0.0f);
    float ty = ((y > 0) ? q[i - Wx] : 0.0f) - ((y < Hy - 1) ? q[i] : 0.0f);
    float tz = ((d > 0) ? s[i - HW] : 0.0f) - ((d < Dz - 1) ? s[i] : 0.0f);
    float tv = t[i];
    float nt = ntv[c];
    tn[i] = tx + ty + tz + tv + nt * (tv - znew[i]);
}

// ---------------------------------------------------------------------------
// WMMA 3x3x3 conv:  out[co,d,h,w] = relu?( sum_tap sum_ci w[co,ci,tap]*in(...) + b[co] )
// Block = 128 threads (4 waves). Each block handles one (d,h) row, 64 x.
// Wave w handles 16 consecutive x as the M dimension of a 16x16x4 f32 WMMA.
// A: lanes 0-15 carry {K0,K1}, lanes 16-31 carry {K2,K3}  (ISA 7.12.2)
// B: lanes 0-15 carry rows {K0,K1}, lanes 16-31 rows {K2,K3}; col N = lane&15
// D: VGPR r -> M = r + 8*(lane>>4), N = lane&15
//
// Weights are zero-padded in LDS to [tap][4][16] so all B reads are
// unconditional (no EXEC save/restore in the WMMA loop), and all 27 B
// fragments are hoisted into VGPRs before the tap loop. Bias is folded into
// the accumulator init; the epilogue stores two float4 per lane.
// ---------------------------------------------------------------------------

__launch_bounds__(128)
__global__ void conv3d_wmma_k(const float* __restrict__ in,   // Cin planes of VOL
                              const float* __restrict__ w,    // [Cout][Cin][27]
                              const float* __restrict__ bias, // [Cout]
                              float* __restrict__ out,        // Cout planes of VOL
                              int Cin, int Cout, int do_relu) {
    __shared__ float tile[4 * 3 * 3 * 66];   // [ci][dz][dy][xx], zero-padded halo
    __shared__ float wpad[27 * 4 * 16];      // [tap][k][n], zero-padded
    __shared__ float blds[16];               // zero-padded bias

    const int tid   = threadIdx.x;
    const int xbase = blockIdx.x * 64;
    const int y     = blockIdx.y;
    const int zz0   = blockIdx.z;

    // zero-padded weights -> LDS  (wpad[tap*64 + k*16 + n])
    for (int i = tid; i < 27 * 64; i += 128) {
        int tap = i >> 6;
        int k   = (i >> 4) & 3;
        int nn  = i & 15;
        float v = 0.0f;
        if (k < Cin && nn < Cout) v = w[(nn * Cin + k) * 27 + tap];
        wpad[i] = v;
    }
    if (tid < 16) blds[tid] = (tid < Cout) ? bias[tid] : 0.0f;

    // input tile -> LDS (zero for OOB and for channels >= Cin)
    for (int i = tid; i < 4 * 3 * 3 * 66; i += 128) {
        int xx = i % 66;
        int r  = i / 66;
        int yy = r % 3;  r /= 3;
        int dz = r % 3;
        int ci = r / 3;
        int gx = xbase + xx - 1;
        int gy = y + yy - 1;
        int gz = zz0 + dz - 1;
        float v = 0.0f;
        if (ci < Cin && gx >= 0 && gx < Wx && gy >= 0 && gy < Hy && gz >= 0 && gz < Dz)
            v = in[ci * VOL + (gz * Hy + gy) * Wx + gx];
        tile[i] = v;
    }
    __syncthreads();

    const int lane = tid & 31;
    const int wave = tid >> 5;
    const int half = lane >> 4;     // 0: lanes 0-15, 1: lanes 16-31
    const int m    = lane & 15;     // M row for A fetch
    const int n    = m;             // N column for B/D
    const int ka   = half * 2;      // A carries input channels ka, ka+1
    const int k0   = half * 2;      // B carries weight rows k0, k0+1

    // preload all 27 B fragments into VGPRs (unconditional ds loads)
    v2f bfrag[27];
#pragma unroll
    for (int tap = 0; tap < 27; ++tap) {
        bfrag[tap].x = wpad[tap * 64 + (k0    ) * 16 + n];
        bfrag[tap].y = wpad[tap * 64 + (k0 + 1) * 16 + n];
    }

    // accumulator starts at bias[n] (0 for padded columns)
    float bsv = blds[n];
    v8f acc;
#pragma unroll
    for (int rr = 0; rr < 8; ++rr) acc[rr] = bsv;

#pragma unroll
    for (int tap = 0; tap < 27; ++tap) {
        int dx  = tap % 3;
        int r   = tap / 3;
        int dy  = r % 3;
        int dzt = r / 3;
        int xx  = wave * 16 + m + dx;   // tile x index (halo built in)

        v2f a;
        a.x = tile[(((ka    ) * 3 + dzt) * 3 + dy) * 66 + xx];
        a.y = tile[(((ka + 1) * 3 + dzt) * 3 + dy) * 66 + xx];

        acc = __builtin_amdgcn_wmma_f32_16x16x4_f32(
            /*neg_a=*/false, a, /*neg_b=*/false, bfrag[tap],
            /*c_mod=*/(short)0, acc, /*reuse_a=*/false, /*reuse_b=*/false);
    }

    // epilogue: optional relu, two float4 stores per valid column
    if (n < Cout) {
        int xout = xbase + wave * 16 + half * 8;     // 8 consecutive x per lane
        int base = (zz0 * Hy + y) * Wx + xout;       // 32B aligned
        float4 lo, hi;
        lo.x = acc[0]; lo.y = acc[1]; lo.z = acc[2]; lo.w = acc[3];
        hi.x = acc[4]; hi.y = acc[5]; hi.z = acc[6]; hi.w = acc[7];
        if (do_relu) {
            lo.x = fmaxf(lo.x, 0.0f); lo.y = fmaxf(lo.y, 0.0f);
            lo.z = fmaxf(lo.z, 0.0f); lo.w = fmaxf(lo.w, 0.0f);
            hi.x = fmaxf(hi.x, 0.0f); hi.y = fmaxf(hi.y, 0.0f);
            hi.z = fmaxf(hi.z, 0.0f); hi.w = fmaxf(hi.w, 0.0f);
        }
        *(float4*)(out + n * VOL + base)     = lo;
        *(float4*)(out + n * VOL + base + 4) = hi;
    }
}

// ---------------------------------------------------------------------------
// Launcher
// ---------------------------------------------------------------------------

extern "C" void kernel_launch(void* const* d_in, const int* in_sizes, int n_in,
                              void* d_out, int out_size, void* d_ws, size_t ws_size,
                              hipStream_t stream) {
    (void)in_sizes; (void)n_in; (void)out_size; (void)ws_size;

    const float* image = (const float*)d_in[0];
    const float* sino  = (const float*)d_in[1];
    const float* W1    = (const float*)d_in[2];   // [4][4][1][27]  stride 108
    const float* B1    = (const float*)d_in[3];   // [4][4]
    const float* W2    = (const float*)d_in[4];   // [4][4][4][27]  stride 432
    const float* B2    = (const float*)d_in[5];   // [4][4]
    const float* W3    = (const float*)d_in[6];   // [4][1][4][27]  stride 108
    const float* B3    = (const float*)d_in[7];   // [4][1]
    const float* ntx   = (const float*)d_in[8];
    const float* nty   = (const float*)d_in[9];
    const float* ntz   = (const float*)d_in[10];
    const float* ntv   = (const float*)d_in[11];

    float* out = (float*)d_out;   // 4 * VOL: [image, t1, t2, t3]
    float* ws  = (float*)d_ws;

    float* z     = ws;
    float* pn    = ws + 1  * (size_t)VOL;
    float* qn    = ws + 2  * (size_t)VOL;
    float* sn    = ws + 3  * (size_t)VOL;
    float* h1    = ws + 4  * (size_t)VOL;  // 4 planes
    float* h2    = ws + 8  * (size_t)VOL;  // 4 planes
    float* blk   = ws + 12 * (size_t)VOL;
    float* p     = ws + 13 * (size_t)VOL;
    float* q     = ws + 14 * (size_t)VOL;
    float* s     = ws + 15 * (size_t)VOL;
    float* resid = ws + 16 * (size_t)VOL;

    dim3 eb(256);
    dim3 eg((VOL + 255) / 256);
    dim3 cb(128);
    dim3 cg(Wx / 64, Hy, Dz);

    copy_k<<<eg, eb, 0, stream>>>(out, image, VOL);
    zero_k<<<dim3((3 * VOL + 255) / 256), eb, 0, stream>>>(p, 3 * VOL);  // p,q,s contiguous

    for (int c = 0; c < NCASC; ++c) {
        const float* t  = out + (size_t)c * VOL;
        float*       tn = out + (size_t)(c + 1) * VOL;

        resid_k<<<dim3((HW + 255) / 256), eb, 0, stream>>>(t, sino, resid);
        z_k<<<eg, eb, 0, stream>>>(t, resid, z);
        diff_k<<<eg, eb, 0, stream>>>(z, pn, qn, sn);

        const float* ins[3]  = { pn, qn, sn };
        float*       accs[3] = { p,  q,  s  };
        const float* nts[3]  = { ntx, nty, ntz };

        for (int j = 0; j < 3; ++j) {
            int bi = j;   // blocks 0,1,2 for p,q,s
            conv3d_wmma_k<<<cg, cb, 0, stream>>>(ins[j], W1 + bi * 108, B1 + bi * 4, h1, 1, 4, 1);
            conv3d_wmma_k<<<cg, cb, 0, stream>>>(h1,     W2 + bi * 432, B2 + bi * 4, h2, 4, 4, 1);
            conv3d_wmma_k<<<cg, cb, 0, stream>>>(h2,     W3 + bi * 108, B3 + bi * 1, blk, 4, 1, 0);
            upd_k<<<eg, eb, 0, stream>>>(accs[j], blk, ins[j], nts[j], c);
        }

        // block 3 on z -> znew (in blk)
        conv3d_wmma_k<<<cg, cb, 0, stream>>>(z,  W1 + 3 * 108, B1 + 3 * 4, h1, 1, 4, 1);
        conv3d_wmma_k<<<cg, cb, 0, stream>>>(h1, W2 + 3 * 432, B2 + 3 * 4, h2, 4, 4, 1);
        conv3d_wmma_k<<<cg, cb, 0, stream>>>(h2, W3 + 3 * 108, B3 + 3 * 1, blk, 4, 1, 0);

        combine_k<<<eg, eb, 0, stream>>>(t, p, q, s, blk, ntv, c, tn);
    }
}